// LocalNorm2d_171798692189
// MI455X (gfx1250) — compile-verified
//
#include <hip/hip_runtime.h>
#include <math.h>

// LocalNorm2d: 33x33 reflect-padded box-filter normalization, fp32,
// shape (32,3,512,512). Memory-bound: staged via CDNA5 async global->LDS
// loads (ASYNCcnt), separable sliding-window box sums in LDS, non-temporal
// output stores to preserve L2 residency of the (re-read) input.

#define HH       512
#define WW       512
#define KS       33
#define PAD      16
#define TILE_H   16
#define TILE_W   128
#define HALO_H   (TILE_H + 2 * PAD)   // 48
#define HALO_W   (TILE_W + 2 * PAD)   // 160
#define XS_STR   168                  // padded LDS stride for x tile
#define CS_STR   161                  // padded LDS stride for column sums
#define NTHREADS 256
#define CW       8                    // output columns per thread (horiz pass)

__device__ __forceinline__ int reflect_idx(int i, int n) {
    i = (i < 0) ? -i : i;             // reflect (exclusive of edge), single bounce
    i = (i >= n) ? (2 * n - 2 - i) : i;
    return i;
}

#if defined(__gfx1250__)
typedef __attribute__((address_space(1))) int   as1_int;
typedef __attribute__((address_space(1))) void  as1_void;
typedef __attribute__((address_space(3))) int   as3_int;
typedef __attribute__((address_space(3))) void  as3_void;
#endif

__device__ __forceinline__ void async_copy_f32(const float* gp, float* lp) {
#if defined(__gfx1250__)
# if __has_builtin(__builtin_amdgcn_global_load_async_to_lds_b32)
    as1_int* g = (as1_int*)(as1_void*)(void*)gp;
    as3_int* l = (as3_int*)(as3_void*)(void*)lp;
    __builtin_amdgcn_global_load_async_to_lds_b32(g, l, /*offset=*/0, /*cpol=*/0);
# else
#  error "gfx1250 device pass: __builtin_amdgcn_global_load_async_to_lds_b32 not available"
# endif
#else
    *lp = *gp;  // host pass placeholder
#endif
}

__device__ __forceinline__ void wait_async_zero() {
#if defined(__gfx1250__)
# if __has_builtin(__builtin_amdgcn_s_wait_asynccnt)
    __builtin_amdgcn_s_wait_asynccnt(0);
# else
    asm volatile("s_wait_asynccnt 0" ::: "memory");
# endif
#endif
}

__global__ __launch_bounds__(NTHREADS)
void LocalNorm2d_kernel(const float* __restrict__ x, float* __restrict__ out) {
    __shared__ float xs[HALO_H * XS_STR];     // 48*168*4 = 32256 B
    __shared__ float colS[TILE_H * CS_STR];   // 16*161*4 = 10304 B
    __shared__ float colQ[TILE_H * CS_STR];   // 10304 B
    __shared__ float outT[TILE_H * TILE_W];   // 8192 B   (total ~61 KB)

    const int tid   = threadIdx.x;
    const int tileX = blockIdx.x;   // 0..3
    const int tileY = blockIdx.y;   // 0..31
    const int plane = blockIdx.z;   // 0..95
    const int gx0 = tileX * TILE_W - PAD;
    const int gy0 = tileY * TILE_H - PAD;
    const float* xp = x + (size_t)plane * (HH * WW);

    // ---- Stage 1: async halo load global -> LDS (reflect-indexed) ----
    #pragma unroll
    for (int i = 0; i < (HALO_H * HALO_W) / NTHREADS; ++i) {   // 30 iters
        int linear = tid + i * NTHREADS;
        int r = linear / HALO_W;
        int c = linear - r * HALO_W;
        int gr = reflect_idx(gy0 + r, HH);
        int gc = reflect_idx(gx0 + c, WW);
        async_copy_f32(xp + gr * WW + gc, &xs[r * XS_STR + c]);
    }
    wait_async_zero();
    __syncthreads();

    // ---- Stage 2: vertical sliding 33-window sums per halo column ----
    if (tid < HALO_W) {
        const int c = tid;
        float s = 0.0f, q = 0.0f;
        #pragma unroll
        for (int k = 0; k < KS; ++k) {
            float v = xs[k * XS_STR + c];
            s += v; q += v * v;
        }
        colS[c] = s; colQ[c] = q;
        #pragma unroll
        for (int r = 1; r < TILE_H; ++r) {
            float vn = xs[(r + KS - 1) * XS_STR + c];
            float vo = xs[(r - 1) * XS_STR + c];
            s += vn - vo;
            q += vn * vn - vo * vo;
            colS[r * CS_STR + c] = s;
            colQ[r * CS_STR + c] = q;
        }
    }
    __syncthreads();

    // ---- Stage 3: horizontal sliding window + normalize + clip ----
    {
        const float inv = 1.0f / (float)(KS * KS);
        const int row = tid >> 4;          // 0..15
        const int c0  = (tid & 15) * CW;   // 0..120
        float s = 0.0f, q = 0.0f;
        #pragma unroll
        for (int j = 0; j < KS; ++j) {
            s += colS[row * CS_STR + c0 + j];
            q += colQ[row * CS_STR + c0 + j];
        }
        #pragma unroll
        for (int k = 0; k < CW; ++k) {
            int col = c0 + k;
            float mean  = s * inv;
            float meanq = q * inv;
            float var   = fabsf(meanq - mean * mean);
            float xv    = xs[(row + PAD) * XS_STR + (col + PAD)];
            float o     = (xv - mean) / (sqrtf(var) + 1e-10f);
            o = fminf(6.0f, fmaxf(-6.0f, o));
            outT[row * TILE_W + col] = o;
            if (k + 1 < CW) {
                s += colS[row * CS_STR + col + KS] - colS[row * CS_STR + col];
                q += colQ[row * CS_STR + col + KS] - colQ[row * CS_STR + col];
            }
        }
    }
    __syncthreads();

    // ---- Stage 4: coalesced, non-temporal tile store ----
    // Output is write-once / never re-read; NT hint keeps the 192MB L2 free
    // for the input halos that neighboring tiles re-read (200MB working set
    // would otherwise thrash L2).
    {
        const int gyo = tileY * TILE_H;
        const int gxo = tileX * TILE_W;
        float* op = out + (size_t)plane * (HH * WW);
        #pragma unroll
        for (int i = 0; i < (TILE_H * TILE_W) / NTHREADS; ++i) {   // 8 iters
            int linear = tid + i * NTHREADS;
            int r = linear >> 7;            // /128
            int c = linear & (TILE_W - 1);
            __builtin_nontemporal_store(outT[linear], &op[(gyo + r) * WW + (gxo + c)]);
        }
    }
}

extern "C" void kernel_launch(void* const* d_in, const int* in_sizes, int n_in,
                              void* d_out, int out_size, void* d_ws, size_t ws_size,
                              hipStream_t stream) {
    (void)n_in; (void)d_ws; (void)ws_size; (void)out_size;
    const float* x = (const float*)d_in[0];
    float* out = (float*)d_out;
    const int planes = in_sizes[0] / (HH * WW);       // 32*3 = 96
    dim3 grid(WW / TILE_W, HH / TILE_H, planes);      // (4, 32, 96)
    LocalNorm2d_kernel<<<grid, dim3(NTHREADS), 0, stream>>>(x, out);
}